// EGNN_36275293782479
// MI455X (gfx1250) — compile-verified
//
#include <hip/hip_runtime.h>
#include <math.h>

#define E_EDGES  400000
#define NN_NODES 25000
#define NG       250

typedef float v2f __attribute__((ext_vector_type(2)));
typedef float v8f __attribute__((ext_vector_type(8)));

// ---------------- workspace layout (float offsets) ----------------
#define CG_F     0          // 64 triples * 343 floats = 21952
#define META_F   22016      // int region (2451 ints used, 4096 reserved)
#define NODE0_F  26112      // 25000*16
#define NODE1_F  426112     // 25000*160
#define NODE2_F  4426112    // 25000*160
#define NODE3_F  8426112    // 25000*64
// total end: 10,026,112 floats (~40.1 MB)

// ---------------- complex double helpers ----------------
struct Cx { double re, im; };
__device__ inline Cx cmul(Cx a, Cx b){ return {a.re*b.re - a.im*b.im, a.re*b.im + a.im*b.re}; }

// Racah formula for integer-l SU(2) Clebsch-Gordan
__device__ double racah_cg(int j1,int m1,int j2,int m2,int j3,int m3){
  double f[12]; f[0]=1.0;
  for (int n=1;n<12;n++) f[n]=f[n-1]*(double)n;
  double pref = sqrt((2.0*j3+1.0)*f[j1+j2-j3]*f[j1-j2+j3]*f[-j1+j2+j3]/f[j1+j2+j3+1]);
  pref *= sqrt(f[j1+m1]*f[j1-m1]*f[j2+m2]*f[j2-m2]*f[j3+m3]*f[j3-m3]);
  int k0 = 0; if (j2-j3-m1>k0) k0=j2-j3-m1; if (j1+m2-j3>k0) k0=j1+m2-j3;
  int k1 = j1+j2-j3; if (j1-m1<k1) k1=j1-m1; if (j2+m2<k1) k1=j2+m2;
  double s=0.0;
  for (int k=k0;k<=k1;k++)
    s += ((k&1)?-1.0:1.0)/(f[k]*f[j1+j2-j3-k]*f[j1-m1-k]*f[j2+m2-k]*f[j3-j2+m1+k]*f[j3-j1-m2+k]);
  return pref*s;
}

// e3nn real->complex change of basis, times (-i)^l
__device__ void fillQ(Cx q[7][7], int l){
  for (int r=0;r<7;r++) for (int c=0;c<7;c++) q[r][c]={0.0,0.0};
  const double is2 = 0.70710678118654752440;
  for (int m=-l;m<0;m++){ q[l+m][l-m]={is2,0.0}; q[l+m][l+m]={0.0,-is2}; }
  q[l][l]={1.0,0.0};
  for (int m=1;m<=l;m++){ double sg=(m&1)?-1.0:1.0; q[l+m][l+m]={sg*is2,0.0}; q[l+m][l-m]={0.0,sg*is2}; }
  int lm=l&3; Cx ph = (lm==0)?Cx{1,0}:(lm==1)?Cx{0,-1}:(lm==2)?Cx{-1,0}:Cx{0,1};
  for (int r=0;r<7;r++) for (int c=0;c<7;c++) q[r][c]=cmul(ph,q[r][c]);
}

// ---------------- TP path enumeration (mirrors _build_tp) ----------------
struct IrrList { int n; int mul[8], l[8], p[8], off[8]; };
__device__ void fill_sh_ir(IrrList& L){ int o=0; L.n=4; for(int l=0;l<4;l++){ L.mul[l]=1; L.l[l]=l; L.p[l]=(l&1)?-1:1; L.off[l]=o; o+=2*l+1; } }
__device__ void fill_mid_ir(IrrList& L){ int o=0,i=0; L.n=8; for(int l=0;l<4;l++) for(int pp=0;pp<2;pp++){ L.mul[i]=5; L.l[i]=l; L.p[i]=pp?1:-1; L.off[i]=o; o+=5*(2*l+1); i++; } }
__device__ void fill_out_ir(IrrList& L){ L.n=1; L.mul[0]=64; L.l[0]=0; L.p[0]=1; L.off[0]=0; }

// record: {o1,m1,d1,o2,d2,oo,mo,dd,wof,trip,pw_bits,pad}
__device__ void emit_tp(int which, int* meta){
  IrrList A,B,O;
  if (which==0) fill_sh_ir(A); else fill_mid_ir(A);
  fill_sh_ir(B);
  if (which==2) fill_out_ir(O); else fill_mid_ir(O);
  int fan[8]; for (int i=0;i<8;i++) fan[i]=0;
  for (int i1=0;i1<A.n;i1++) for (int i2=0;i2<B.n;i2++) for (int io=0;io<O.n;io++){
    int dl=A.l[i1]-B.l[i2]; if (dl<0) dl=-dl;
    if (O.p[io]==A.p[i1]*B.p[i2] && dl<=O.l[io] && O.l[io]<=A.l[i1]+B.l[i2])
      fan[io]+=A.mul[i1]*B.mul[i2];
  }
  int cnt=0, wof=0; int* rec = meta + 3 + which*68*12;
  for (int i1=0;i1<A.n;i1++) for (int i2=0;i2<B.n;i2++) for (int io=0;io<O.n;io++){
    int l1=A.l[i1], l2=B.l[i2], lo=O.l[io];
    int dl=l1-l2; if (dl<0) dl=-dl;
    if (!(O.p[io]==A.p[i1]*B.p[i2] && dl<=lo && lo<=l1+l2)) continue;
    rec[0]=A.off[i1]; rec[1]=A.mul[i1]; rec[2]=2*l1+1;
    rec[3]=B.off[i2]; rec[4]=2*l2+1;
    rec[5]=O.off[io]; rec[6]=O.mul[io]; rec[7]=2*lo+1;
    rec[8]=wof; rec[9]=(l1*4+l2)*4+lo;
    float pw = sqrtf((float)(2*lo+1)/(float)fan[io]);
    rec[10]=__float_as_int(pw); rec[11]=0;
    wof += A.mul[i1]*B.mul[i2]*O.mul[io];
    rec+=12; cnt++;
  }
  meta[which]=cnt;
}

// ---------------- setup: Wigner-3j tables + path metadata ----------------
__global__ void setup_kernel(float* cg, int* meta){
  int b = blockIdx.x;                 // b = (l1<<4)|(l2<<2)|l3
  int l1=(b>>4)&3, l2=(b>>2)&3, l3=b&3;
  int tid = threadIdx.x;
  int dl=l1-l2; if (dl<0) dl=-dl;
  bool valid = (l3>=dl) && (l3<=l1+l2);
  __shared__ Cx Q1[7][7], Q2[7][7], Q3[7][7];
  __shared__ double Csu2[343], Cre[343], red[256];
  if (valid){
    int d1=2*l1+1, d2=2*l2+1, d3=2*l3+1, tot=d1*d2*d3;
    if (tid==0){ fillQ(Q1,l1); fillQ(Q2,l2); fillQ(Q3,l3); }
    __syncthreads();
    for (int idx=tid; idx<tot; idx+=blockDim.x){
      int i=idx/(d2*d3), r=idx%(d2*d3), k=r/d3, n=r%d3;
      int m1=i-l1, m2=k-l2, m3=n-l3;
      Csu2[idx] = (m1+m2==m3)? racah_cg(l1,m1,l2,m2,l3,m3) : 0.0;
    }
    __syncthreads();
    for (int idx=tid; idx<tot; idx+=blockDim.x){
      int j=idx/(d2*d3), r=idx%(d2*d3), l=r/d3, m=r%d3;
      double sre=0.0;
      for (int i=0;i<d1;i++){ Cx a=Q1[i][j];
        for (int k=0;k<d2;k++){ Cx ab=cmul(a,Q2[k][l]);
          for (int n=0;n<d3;n++){
            double cv=Csu2[(i*d2+k)*d3+n]; if (cv==0.0) continue;
            Cx c3; c3.re=Q3[n][m].re; c3.im=-Q3[n][m].im;
            Cx t=cmul(ab,c3);
            sre += t.re*cv;
          } } }
      Cre[idx]=sre;
    }
    __syncthreads();
    double s=0.0;
    for (int idx=tid; idx<tot; idx+=blockDim.x) s+=Cre[idx]*Cre[idx];
    red[tid]=s; __syncthreads();
    for (int off=blockDim.x>>1; off>0; off>>=1){ if (tid<(unsigned)off) red[tid]+=red[tid+off]; __syncthreads(); }
    double nrm = sqrt(red[0]);
    double rn = (nrm>0.0)? 1.0/nrm : 0.0;
    float* T = cg + b*343;
    for (int idx=tid; idx<tot; idx+=blockDim.x) T[idx]=(float)(Cre[idx]*rn);
  }
  if (b==0 && tid==0){ emit_tp(0,meta); emit_tp(1,meta); emit_tp(2,meta); }
}

// ---------------- spherical harmonics lmax=3 (component norm) ----------------
__device__ inline void compute_sh(const float* __restrict__ v, float* sh){
  float x=v[0], y=v[1], z=v[2];
  const float s3=1.7320508075688772f, s5=2.2360679774997896f, s7=2.6457513110645907f;
  float y2=y*y, x2z2=x*x+z*z;
  float sh20=s3*x*z, sh21=s3*x*y, sh22=y2-0.5f*x2z2, sh23=s3*y*z, sh24=0.5f*s3*(z*z-x*x);
  const float c30=0.9128709291752769f;   // sqrt(5/6)
  const float c32=0.6123724356957945f;   // sqrt(3/8)
  sh[0]=1.f; sh[1]=s3*x; sh[2]=s3*y; sh[3]=s3*z;
  sh[4]=s5*sh20; sh[5]=s5*sh21; sh[6]=s5*sh22; sh[7]=s5*sh23; sh[8]=s5*sh24;
  sh[9] =s7*(c30*(sh20*z+sh24*x));
  sh[10]=s7*(s5*sh20*y);
  sh[11]=s7*(c32*(4.f*y2-x2z2)*x);
  sh[12]=s7*(0.5f*y*(2.f*y2-3.f*x2z2));
  sh[13]=s7*(c32*z*(4.f*y2-x2z2));
  sh[14]=s7*(s5*sh24*y);
  sh[15]=s7*(c30*(sh24*z-sh20*x));
}

// ---------------- zero fill ----------------
__global__ void zero_kernel(float* p, int n){
  for (int i = blockIdx.x*blockDim.x + threadIdx.x; i < n; i += gridDim.x*blockDim.x) p[i]=0.f;
}

// ---------------- pass 0: sh + scatter ----------------
__global__ void sh_scatter_kernel(const float* __restrict__ ev, const int* __restrict__ dstI,
                                  float* __restrict__ node0, float inv){
  int e = blockIdx.x*blockDim.x + threadIdx.x;
  if (e >= E_EDGES) return;
  float sh[16]; compute_sh(ev + 3*(size_t)e, sh);
  float* row = node0 + (size_t)dstI[e]*16;
  #pragma unroll
  for (int i=0;i<16;i++) atomicAdd(row+i, sh[i]*inv);
}

// ---------------- generic fused TP + scatter (TP_IN / TP_LAYER) ----------------
#define TPB_TP 64
__global__ void tp_kernel(const float* __restrict__ nodeIn, float* __restrict__ nodeOut,
                          const int* __restrict__ srcI, const int* __restrict__ dstI,
                          const float* __restrict__ ev, const float* __restrict__ w,
                          const float* __restrict__ cg, const int* __restrict__ meta,
                          int which, int d1tot, int dout, float inv){
  extern __shared__ float smem[];               // [dout][TPB_TP], conflict-free per-thread acc
  int tid = threadIdx.x;
  int e = blockIdx.x*TPB_TP + tid;              // grid sized exactly: no tail
  for (int i=0;i<dout;i++) smem[i*TPB_TP+tid]=0.f;
  float shv[16]; compute_sh(ev + 3*(size_t)e, shv);
  const float* arow = nodeIn + (size_t)srcI[e]*d1tot;
  int np = meta[which];
  const int* rec = meta + 3 + which*68*12;
  for (int p=0;p<np;p++){
    const int* r = rec + p*12;
    int o1=r[0],m1=r[1],d1=r[2],o2=r[3],d2=r[4],oo=r[5],mo=r[6],dd=r[7],wof=r[8],trip=r[9];
    float pw = __int_as_float(r[10]);
    const float* w3 = cg + trip*343;
    float t[5][7];
    #pragma unroll
    for (int u=0;u<5;u++){
      #pragma unroll
      for (int k=0;k<7;k++) t[u][k]=0.f;
    }
    for (int i=0;i<d1;i++){
      float av[5];
      #pragma unroll
      for (int u=0;u<5;u++) av[u] = (u<m1) ? arow[o1+u*d1+i] : 0.f;
      for (int j=0;j<d2;j++){
        float bv = shv[o2+j];
        const float* w3p = w3 + (i*d2+j)*dd;
        #pragma unroll
        for (int k=0;k<7;k++){
          if (k<dd){
            float cw = bv * w3p[k];               // wave-uniform CG -> scalar loads
            #pragma unroll
            for (int u=0;u<5;u++) if (u<m1) t[u][k] += av[u]*cw;
          }
        }
      }
    }
    for (int wo=0; wo<mo; wo++){
      float wu[5];
      #pragma unroll
      for (int u=0;u<5;u++) wu[u] = (u<m1) ? w[wof + u*mo + wo] : 0.f;
      #pragma unroll
      for (int k=0;k<7;k++){
        if (k<dd){
          float s=0.f;
          #pragma unroll
          for (int u=0;u<5;u++) if (u<m1) s += t[u][k]*wu[u];
          smem[(oo + wo*dd + k)*TPB_TP + tid] += pw*s;
        }
      }
    }
  }
  float* orow = nodeOut + (size_t)dstI[e]*dout;
  for (int i=0;i<dout;i++) atomicAdd(orow+i, inv*smem[i*TPB_TP+tid]);
}

// ---------------- TP_OUT as WMMA GEMM: [16 edges x 20] x [20 x 64] ----------------
__global__ void tp_out_wmma_kernel(const float* __restrict__ nodeIn, float* __restrict__ nodeOut,
                                   const int* __restrict__ srcI, const int* __restrict__ dstI,
                                   const float* __restrict__ ev, const float* __restrict__ w_out,
                                   const float* __restrict__ cg, const int* __restrict__ meta,
                                   float inv){
  __shared__ float Atile[8][16][20];            // per-wave A tile
  int wave = threadIdx.x>>5, lane = threadIdx.x&31;
  int tile = blockIdx.x*8 + wave;
  int e0 = tile*16;
  int np = meta[2];                              // = 4 paths (l=0..3, lo=0)
  const int* rec = meta + 3 + 2*68*12;
  if (lane < 16){
    int e = e0 + lane;
    float shv[16]; compute_sh(ev + 3*(size_t)e, shv);
    const float* arow = nodeIn + (size_t)srcI[e]*160;
    for (int p=0;p<np;p++){
      const int* r = rec + p*12;
      int o1=r[0], d1=r[2], o2=r[3], d2=r[4], trip=r[9];  // dd==1
      float pw = __int_as_float(r[10]);
      const float* w3 = cg + trip*343;
      #pragma unroll
      for (int u=0;u<5;u++){
        float s=0.f;
        for (int i=0;i<d1;i++){
          float av = arow[o1+u*d1+i];
          for (int j=0;j<d2;j++) s += av*shv[o2+j]*w3[i*d2+j];
        }
        Atile[wave][lane][p*5+u] = pw*s;
      }
    }
  }
  __syncthreads();
  int row = lane & 15, khalf = lane >> 4;        // f32 A frag: M=lane%16, K=r+2*(lane/16)
  v8f c[4];
  #pragma unroll
  for (int nt=0;nt<4;nt++){
    #pragma unroll
    for (int q=0;q<8;q++) c[nt][q]=0.f;
  }
  #pragma unroll
  for (int kk=0; kk<5; kk++){                    // K = 20 in 5 steps of 4
    v2f a;
    a.x = Atile[wave][row][kk*4 + 0 + 2*khalf];
    a.y = Atile[wave][row][kk*4 + 1 + 2*khalf];
    int kbase = kk*4 + 2*khalf;
    #pragma unroll
    for (int nt=0; nt<4; nt++){                  // N = 64 in 4 tiles of 16
      int col = nt*16 + row;
      v2f b;
      b.x = w_out[(kbase+0)*64 + col];           // W2[lu][w] == w_out flat (wof = lu*64)
      b.y = w_out[(kbase+1)*64 + col];
      c[nt] = __builtin_amdgcn_wmma_f32_16x16x4_f32(false, a, false, b, (short)0, c[nt], false, false);
    }
  }
  // D layout: VGPR q, lanes0-15 -> M=q; lanes16-31 -> M=q+8; N = lane%16
  #pragma unroll
  for (int nt=0; nt<4; nt++){
    int col = nt*16 + row;
    #pragma unroll
    for (int q=0; q<8; q++){
      int m = q + 8*khalf;
      int e = e0 + m;
      int d = dstI[e];
      atomicAdd(nodeOut + (size_t)d*64 + col, c[nt][q]*inv);
    }
  }
}

// ---------------- graph pool ----------------
__global__ void graph_kernel(const float* __restrict__ node3, const int* __restrict__ batch,
                             float* __restrict__ out){
  int idx = blockIdx.x*blockDim.x + threadIdx.x;     // exact: 25000*64
  if (idx >= NN_NODES*64) return;
  int n = idx>>6, col = idx&63;
  atomicAdd(out + batch[n]*64 + col, node3[idx]*1e-5f);   // 1/(sqrt(100)*10000)
}

// ---------------- launcher ----------------
extern "C" void kernel_launch(void* const* d_in, const int* in_sizes, int n_in,
                              void* d_out, int out_size, void* d_ws, size_t ws_size,
                              hipStream_t stream){
  const int*   edge_index = (const int*)d_in[0];     // [2, E]
  const float* edge_vec   = (const float*)d_in[1];   // [E, 3]
  const int*   batch      = (const int*)d_in[2];     // [N]
  const float* w_in       = (const float*)d_in[3];
  const float* w_layer0   = (const float*)d_in[4];
  const float* w_out      = (const float*)d_in[5];
  const int* src = edge_index;
  const int* dst = edge_index + E_EDGES;

  float* ws    = (float*)d_ws;
  float* cg    = ws + CG_F;
  int*   meta  = (int*)(ws + META_F);
  float* node0 = ws + NODE0_F;
  float* node1 = ws + NODE1_F;
  float* node2 = ws + NODE2_F;
  float* node3 = ws + NODE3_F;
  float* out   = (float*)d_out;
  const float inv = 0.25f;                           // 1/sqrt(NUM_NEIGHBORS)

  setup_kernel<<<64, 256, 0, stream>>>(cg, meta);
  zero_kernel<<<2048, 256, 0, stream>>>(node0, 400000 + 4000000 + 4000000 + 1600000);
  zero_kernel<<<64, 256, 0, stream>>>(out, NG*64);
  sh_scatter_kernel<<<(E_EDGES+255)/256, 256, 0, stream>>>(edge_vec, dst, node0, inv);
  tp_kernel<<<E_EDGES/TPB_TP, TPB_TP, 160*TPB_TP*4, stream>>>
      (node0, node1, src, dst, edge_vec, w_in,     cg, meta, 0,  16, 160, inv);
  tp_kernel<<<E_EDGES/TPB_TP, TPB_TP, 160*TPB_TP*4, stream>>>
      (node1, node2, src, dst, edge_vec, w_layer0, cg, meta, 1, 160, 160, inv);
  tp_out_wmma_kernel<<<E_EDGES/(16*8), 256, 0, stream>>>
      (node2, node3, src, dst, edge_vec, w_out, cg, meta, inv);
  graph_kernel<<<(NN_NODES*64)/256, 256, 0, stream>>>(node3, batch, out);
}